// GCN_75642964017551
// MI455X (gfx1250) — compile-verified
//
#include <hip/hip_runtime.h>
#include <hip/hip_bf16.h>
#include <stdint.h>

typedef float v2f __attribute__((ext_vector_type(2)));
typedef float v8f __attribute__((ext_vector_type(8)));

#define GEMM_K 128           // inner dim is 128 for all three layers
#define ROW_TILES 8          // 16-row tiles per wave -> 128 rows per block

// ----------------------------------------------------------------------------
// Edge index conversion: int64 [2,E] -> two int32 arrays (N < 2^31).
// ----------------------------------------------------------------------------
__global__ void k_cvt_edges(const long long* __restrict__ ei,
                            int* __restrict__ src32, int* __restrict__ dst32,
                            unsigned e_cnt) {
  unsigned i = blockIdx.x * blockDim.x + threadIdx.x;
  if (i < e_cnt) {
    src32[i] = (int)ei[i];
    dst32[i] = (int)ei[(size_t)e_cnt + i];
  }
}

// ----------------------------------------------------------------------------
// Degree / normalization precompute
// ----------------------------------------------------------------------------
__global__ void k_init_deg(float* __restrict__ deg, int n) {
  int i = blockIdx.x * blockDim.x + threadIdx.x;
  if (i < n) deg[i] = 1.0f;  // self-loop contributes weight 1
}

__global__ void k_accum_deg(const int* __restrict__ dst,
                            const float* __restrict__ ew,
                            float* __restrict__ deg, unsigned e_cnt) {
  unsigned i = blockIdx.x * blockDim.x + threadIdx.x;
  if (i < e_cnt) atomicAdd(&deg[dst[i]], ew[i]);
}

__global__ void k_dinv(float* __restrict__ deg, int n) {
  int i = blockIdx.x * blockDim.x + threadIdx.x;
  if (i < n) {
    float d = deg[i];
    deg[i] = (d > 0.0f) ? rsqrtf(d) : 0.0f;  // overwrite in place with deg^-1/2
  }
}

__global__ void k_edge_norm(const int* __restrict__ src,
                            const int* __restrict__ dst,
                            const float* __restrict__ ew,
                            const float* __restrict__ dinv,
                            float* __restrict__ norm, unsigned e_cnt) {
  unsigned i = blockIdx.x * blockDim.x + threadIdx.x;
  if (i < e_cnt) norm[i] = dinv[src[i]] * ew[i] * dinv[dst[i]];
}

// ----------------------------------------------------------------------------
// fp32 GEMM via V_WMMA_F32_16X16X4_F32, K fixed at 128.
// blockDim.x = 32 * (OUT/16): wave w owns output columns [16w, 16w+16).
// Each wave preloads its 16-column slice of W into 64 VGPRs (v2f bfrag[32],
// per the 4x16 f32 B layout: lane<16 holds K=k0..k0+1, lane>=16 K=k0+2..k0+3
// for column n0+lane%16), then sweeps ROW_TILES 16-row tiles, so per tile the
// inner loop issues only 32 independent global_load_b64 of A (immediate
// offsets off one base) feeding the chained WMMAs.
// D layout: VGPR j -> row m0 + j + 8*(lane/16), col n0 + lane%16.
// ----------------------------------------------------------------------------
__global__ void __launch_bounds__(256)
k_gemm_wmma(const float* __restrict__ X,
            const float* __restrict__ W,
            float* __restrict__ H,
            int n, int OUT) {
  const int lane = threadIdx.x & 31;
  const int wave = threadIdx.x >> 5;
  const int half = lane >> 4;   // 0: lanes 0-15, 1: lanes 16-31
  const int lid  = lane & 15;
  const int n0   = wave * 16;
  const int mbase = blockIdx.x * (16 * ROW_TILES);

  // --- preload B fragments for this wave's column slice (64 VGPRs) ---
  v2f bfrag[GEMM_K / 4];
  {
    const float* __restrict__ bcol = W + n0 + lid;
#pragma unroll
    for (int kk = 0; kk < GEMM_K / 4; ++kk) {
      const int ka = 4 * kk + 2 * half;
      bfrag[kk].x = bcol[(unsigned)ka * (unsigned)OUT];
      bfrag[kk].y = bcol[(unsigned)(ka + 1) * (unsigned)OUT];
    }
  }

  // --- sweep row tiles ---
  for (int mt = 0; mt < ROW_TILES; ++mt) {
    const int m0 = mbase + mt * 16;
    if (m0 >= n) break;
    int ar = m0 + lid;
    if (ar >= n) ar = n - 1;                    // clamp duplicate (tail rows)
    const float* __restrict__ arow =
        X + (unsigned)ar * (unsigned)GEMM_K + 2 * half;

    v8f acc = {};
#pragma unroll
    for (int kk = 0; kk < GEMM_K / 4; ++kk) {
      v2f a = *(const v2f*)(arow + 4 * kk);     // A[M=lid][ka], A[M=lid][ka+1]
      acc = __builtin_amdgcn_wmma_f32_16x16x4_f32(
          /*neg_a=*/false, a, /*neg_b=*/false, bfrag[kk],
          /*c_mod=*/(short)0, acc, /*reuse_a=*/false, /*reuse_b=*/false);
    }

    const int row0 = m0 + 8 * half;
    float* __restrict__ hp = H + (unsigned)row0 * (unsigned)OUT + n0 + lid;
#pragma unroll
    for (int j = 0; j < 8; ++j)
      if (row0 + j < n) hp[(unsigned)j * (unsigned)OUT] = acc[j];
  }
}

// ----------------------------------------------------------------------------
// out[v, 4q..4q+3] = H[v, ...] * dinv[v]^2 + bias[...]   (float4 per thread;
// fully initializes the output buffer before the edge scatter-adds)
// ----------------------------------------------------------------------------
__global__ void k_self_bias(const float* __restrict__ H,
                            const float* __restrict__ dinv,
                            const float* __restrict__ bias,
                            float* __restrict__ out,
                            int n, int fshift) {
  const int qshift = fshift - 2;
  const unsigned total = (unsigned)n << qshift;
  unsigned idx = blockIdx.x * blockDim.x + threadIdx.x;
  if (idx >= total) return;
  const unsigned v = idx >> qshift;
  const unsigned q = idx & ((1u << qshift) - 1);
  const float di = dinv[v];
  const float d2 = di * di;
  const float4 h = *(const float4*)(H + (idx << 2));
  const float4 b = *(const float4*)(bias + (q << 2));
  float4 o;
  o.x = h.x * d2 + b.x;
  o.y = h.y * d2 + b.y;
  o.z = h.z * d2 + b.z;
  o.w = h.w * d2 + b.w;
  *(float4*)(out + (idx << 2)) = o;
}

// ----------------------------------------------------------------------------
// Edge scatter: out[dst, f] += H[src, f] * norm[e], 4 features per thread.
// F/4 consecutive threads share one edge -> b128 coalesced gather of H,
// broadcast src/dst/norm loads, f32 L2 atomics for the scatter.
// ----------------------------------------------------------------------------
__global__ void k_scatter(const float* __restrict__ H,
                          const int* __restrict__ src,
                          const int* __restrict__ dst,
                          const float* __restrict__ norm,
                          float* __restrict__ out,
                          unsigned e_cnt, int fshift) {
  const int qshift = fshift - 2;
  const unsigned total = e_cnt << qshift;
  unsigned idx = blockIdx.x * blockDim.x + threadIdx.x;
  if (idx >= total) return;
  const unsigned e = idx >> qshift;
  const unsigned q = idx & ((1u << qshift) - 1);
  const unsigned s = (unsigned)src[e];
  const unsigned d = (unsigned)dst[e];
  const float w = norm[e];
  const float4 h = *(const float4*)(H + (s << fshift) + (q << 2));
  float* __restrict__ op = out + (d << fshift) + (q << 2);
  atomicAdd(op + 0, h.x * w);
  atomicAdd(op + 1, h.y * w);
  atomicAdd(op + 2, h.z * w);
  atomicAdd(op + 3, h.w * w);
}

__global__ void k_relu(float* __restrict__ x, unsigned total_q) {
  unsigned i = blockIdx.x * blockDim.x + threadIdx.x;
  if (i >= total_q) return;
  float4 v = *(const float4*)(x + (i << 2));
  v.x = fmaxf(v.x, 0.0f);
  v.y = fmaxf(v.y, 0.0f);
  v.z = fmaxf(v.z, 0.0f);
  v.w = fmaxf(v.w, 0.0f);
  *(float4*)(x + (i << 2)) = v;
}

// ----------------------------------------------------------------------------
// Orchestration
// ----------------------------------------------------------------------------
extern "C" void kernel_launch(void* const* d_in, const int* in_sizes, int n_in,
                              void* d_out, int out_size, void* d_ws, size_t ws_size,
                              hipStream_t stream) {
  (void)n_in; (void)out_size; (void)ws_size;

  const int D0 = 128, D1 = 128, D2 = 128, D3 = 64;

  const float*     x  = (const float*)d_in[0];
  const long long* ei = (const long long*)d_in[1];   // int64 [2, E]
  const float*     ew = (const float*)d_in[2];
  const float*     W1 = (const float*)d_in[3];
  const float*     b1 = (const float*)d_in[4];
  const float*     W2 = (const float*)d_in[5];
  const float*     b2 = (const float*)d_in[6];
  const float*     W3 = (const float*)d_in[7];
  const float*     b3 = (const float*)d_in[8];

  const int      N = in_sizes[0] / D0;        // 50000
  const unsigned E = (unsigned)in_sizes[2];   // 1600000

  // Workspace layout: dinv[N] f32 | norm[E] f32 | src32[E] i32 | dst32[E] i32
  //                   | bufA[N*128] f32 | bufB[N*128] f32
  float* ws    = (float*)d_ws;
  float* dinv  = ws;
  float* norm  = dinv + N;
  int*   src32 = (int*)(norm + E);
  int*   dst32 = src32 + E;
  float* bufA  = (float*)(dst32 + E);
  float* bufB  = bufA + (size_t)N * 128;

  float* out = (float*)d_out;                 // [N, 64]

  const int BT = 256;
  const int gN = (N + BT - 1) / BT;
  const int gE = (int)((E + BT - 1) / BT);
  const int gG = (N + 16 * ROW_TILES - 1) / (16 * ROW_TILES);  // GEMM blocks

  // --- edge conversion + normalization precompute ---
  k_cvt_edges<<<gE, BT, 0, stream>>>(ei, src32, dst32, E);
  k_init_deg <<<gN, BT, 0, stream>>>(dinv, N);
  k_accum_deg<<<gE, BT, 0, stream>>>(dst32, ew, dinv, E);
  k_dinv     <<<gN, BT, 0, stream>>>(dinv, N);
  k_edge_norm<<<gE, BT, 0, stream>>>(src32, dst32, ew, dinv, norm, E);

  // --- layer 1: x(128) -> bufB(128), relu ---
  k_gemm_wmma<<<gG, 32 * (D1 / 16), 0, stream>>>(x, W1, bufA, N, D1);
  {
    const unsigned nq = (unsigned)N * (D1 / 4);
    const unsigned eq = E << 5;               // E * 128/4
    k_self_bias<<<(int)((nq + BT - 1) / BT), BT, 0, stream>>>(bufA, dinv, b1, bufB, N, 7);
    k_scatter  <<<(int)((eq + BT - 1) / BT), BT, 0, stream>>>(bufA, src32, dst32, norm, bufB, E, 7);
    k_relu     <<<(int)((nq + BT - 1) / BT), BT, 0, stream>>>(bufB, nq);
  }

  // --- layer 2: bufB(128) -> bufB(128), relu (bufA holds H2; bufB rewritten) ---
  k_gemm_wmma<<<gG, 32 * (D2 / 16), 0, stream>>>(bufB, W2, bufA, N, D2);
  {
    const unsigned nq = (unsigned)N * (D2 / 4);
    const unsigned eq = E << 5;
    k_self_bias<<<(int)((nq + BT - 1) / BT), BT, 0, stream>>>(bufA, dinv, b2, bufB, N, 7);
    k_scatter  <<<(int)((eq + BT - 1) / BT), BT, 0, stream>>>(bufA, src32, dst32, norm, bufB, E, 7);
    k_relu     <<<(int)((nq + BT - 1) / BT), BT, 0, stream>>>(bufB, nq);
  }

  // --- layer 3: bufB(128) -> d_out(64), no relu ---
  k_gemm_wmma<<<gG, 32 * (D3 / 16), 0, stream>>>(bufB, W3, bufA, N, D3);
  {
    const unsigned nq = (unsigned)N * (D3 / 4);
    const unsigned eq = E << 4;               // E * 64/4
    k_self_bias<<<(int)((nq + BT - 1) / BT), BT, 0, stream>>>(bufA, dinv, b3, out, N, 6);
    k_scatter  <<<(int)((eq + BT - 1) / BT), BT, 0, stream>>>(bufA, src32, dst32, norm, out, E, 6);
  }
}